// SelfAlignedCrossAttention_34333968564398
// MI455X (gfx1250) — compile-verified
//
#include <hip/hip_runtime.h>
#include <hip/hip_bf16.h>
#include <math.h>

// ---------------------------------------------------------------------------
// SelfAlignedCrossAttention for MI455X (gfx1250, wave32, WMMA f32 16x16x4)
// Outputs (flat, in order): aqq [32,8,512,512], aqs [32,8,512,512],
//                           a   [32,8,512,1024], out [32,512,128]
// ---------------------------------------------------------------------------

typedef __attribute__((ext_vector_type(2))) float v2f;
typedef __attribute__((ext_vector_type(8))) float v8f;

#define BS      32
#define LT      512
#define LSQ     512
#define DQ      128
#define NH      8
#define DMODEL  128
#define NHD     1024   // NH * DQ

// ---- WMMA helpers ---------------------------------------------------------

__device__ __forceinline__ v8f wmma4(v2f a, v2f b, v8f c) {
    // D = A(16x4) * B(4x16) + C(16x16), fp32
    return __builtin_amdgcn_wmma_f32_16x16x4_f32(
        /*neg_a=*/false, a, /*neg_b=*/false, b,
        /*c_mod=*/(short)0, c, /*reuse_a=*/false, /*reuse_b=*/false);
}

// A fragment: A row-major [16][lda] (pre-offset to tile origin).
// lane l holds A[l&15][kb + 2*(l>>4) + {0,1}]  -> contiguous pair
__device__ __forceinline__ v2f load_a_frag(const float* __restrict__ A,
                                           int lda, int kb, int lane) {
    const float* p = A + (size_t)(lane & 15) * lda + kb + 2 * (lane >> 4);
    v2f a; a.x = p[0]; a.y = p[1];
    return a;
}

// B fragment from K-major source: B[k][n], B row-major [K][ldb] (pre-offset to n0)
// lane l holds B[kb + 2*(l>>4) + {0,1}][l&15]  -> strided pair
__device__ __forceinline__ v2f load_b_kn(const float* __restrict__ B,
                                         int ldb, int kb, int lane) {
    int n = lane & 15;
    int k = kb + 2 * (lane >> 4);
    v2f b; b.x = B[(size_t)k * ldb + n]; b.y = B[(size_t)(k + 1) * ldb + n];
    return b;
}

// B fragment where B = X^T and X is row-major [N][ldx] (keys): B[k][n] = X[n][k]
// lane l holds X[l&15][kb + 2*(l>>4) + {0,1}]  -> contiguous pair
__device__ __forceinline__ v2f load_b_nk(const float* __restrict__ X,
                                         int ldx, int kb, int lane) {
    const float* p = X + (size_t)(lane & 15) * ldx + kb + 2 * (lane >> 4);
    v2f b; b.x = p[0]; b.y = p[1];
    return b;
}

// C/D tile store: D row-major [16][ldd] (pre-offset to tile origin).
// vgpr i -> row (i + 8*(l>>4)), col l&15
__device__ __forceinline__ void store_c(float* __restrict__ D, int ldd,
                                        v8f c, int lane) {
    int n  = lane & 15;
    int rb = 8 * (lane >> 4);
#pragma unroll
    for (int i = 0; i < 8; ++i) D[(size_t)(rb + i) * ldd + n] = c[i];
}

// ---- Kernel 1: fused Q / K projections -----------------------------------
// qt[b,h,t,d] = (hla @ W_Q), qs[b,h,s,d] = (pep @ W_K)
// One wave per 16x16 output tile. 131072 waves total.
__global__ __launch_bounds__(256)
void proj_kernel(const float* __restrict__ hla, const float* __restrict__ pep,
                 const float* __restrict__ WQ,  const float* __restrict__ WK,
                 float* __restrict__ qt, float* __restrict__ qs) {
    int lane = threadIdx.x & 31;
    int wave = threadIdx.x >> 5;
    int gw   = blockIdx.x * 8 + wave;          // 0 .. 131071
    int which = gw >> 16;                      // 0 = Q path, 1 = K path
    int tile  = gw & 0xFFFF;                   // 0 .. 65535
    int mtile = tile >> 6;                     // 0 .. 1023  (rows of 16)
    int ntile = tile & 63;                     // 0 .. 63    (cols of 16)

    const float* X = which ? pep : hla;        // [16384][128]
    const float* W = which ? WK  : WQ;         // [128][1024]
    float*       O = which ? qs  : qt;         // [b][h][t][128]

    int m0 = mtile * 16;
    int n0 = ntile * 16;
    const float* A = X + (size_t)m0 * DMODEL;

    v8f c = {};
#pragma unroll 4
    for (int kb = 0; kb < 32; ++kb) {
        v2f a = load_a_frag(A, DMODEL, kb * 4, lane);
        v2f b = load_b_kn(W + n0, NHD, kb * 4, lane);
        c = wmma4(a, b, c);
    }

    // scatter into [b][h][t][d]; a 16x16 tile never crosses b/h boundaries
    int b = m0 >> 9;            // / 512
    int t = m0 & 511;
    int h = n0 >> 7;            // / 128
    int d = n0 & 127;
    float* D = O + (((size_t)(b * NH + h) * LT + t)) * DQ + d;
    store_c(D, DQ, c, lane);
}

// ---- Kernel 2: fused scores + softmax + A*V + residual -------------------
// Grid: (LT/16, BS*NH). Block = 256 threads = 8 waves, LDS = 16x1024 f32.
__global__ __launch_bounds__(256)
void attn_kernel(const float* __restrict__ qt_all, const float* __restrict__ qs_all,
                 float* __restrict__ aqq, float* __restrict__ aqs,
                 float* __restrict__ a_out, float* __restrict__ that) {
    __shared__ float S[16 * 1024];             // 64 KB: scores then probs

    int bh   = blockIdx.y;                     // b*8 + h
    int t0   = blockIdx.x * 16;
    int lane = threadIdx.x & 31;
    int wave = threadIdx.x >> 5;

    const float* Q  = qt_all + (size_t)bh * LT  * DQ + (size_t)t0 * DQ;
    const float* Kq = qt_all + (size_t)bh * LT  * DQ;
    const float* Ks = qs_all + (size_t)bh * LSQ * DQ;

    // Preload the 16x128 Q tile's A-fragments once; reused across 8 col tiles.
    v2f afrag[32];
#pragma unroll
    for (int kb = 0; kb < 32; ++kb) afrag[kb] = load_a_frag(Q, DQ, kb * 4, lane);

    // ---- raw scores: S[16][1024] = Q @ [qt;qs]^T  (no 1/sqrt(d) in reference)
    for (int ct = wave; ct < 64; ct += 8) {
        int s0 = ct * 16;
        const float* Kp = (ct < 32) ? (Kq + (size_t)s0 * DQ)
                                    : (Ks + (size_t)(s0 - LT) * DQ);
        v8f c = {};
#pragma unroll 4
        for (int kb = 0; kb < 32; ++kb) {
            v2f b = load_b_nk(Kp, DQ, kb * 4, lane);
            c = wmma4(afrag[kb], b, c);
        }
        int n  = lane & 15;
        int rb = 8 * (lane >> 4);
        float* g = (ct < 32) ? (aqq + ((size_t)bh * LT + t0) * LT  + s0)
                             : (aqs + ((size_t)bh * LT + t0) * LSQ + (s0 - LT));
#pragma unroll
        for (int i = 0; i < 8; ++i) {
            float v = c[i];
            S[(rb + i) * 1024 + s0 + n] = v;
            g[(size_t)(rb + i) * 512 + n] = v;
        }
    }
    __syncthreads();

    // ---- softmax per row (wave handles 2 rows, shfl_xor reductions)
#pragma unroll
    for (int rr = 0; rr < 2; ++rr) {
        int r = wave * 2 + rr;
        float* row = &S[r * 1024];
        float m = -3.402823466e38f;
        for (int j = lane; j < 1024; j += 32) m = fmaxf(m, row[j]);
#pragma unroll
        for (int off = 16; off > 0; off >>= 1) m = fmaxf(m, __shfl_xor(m, off, 32));
        float sum = 0.f;
        for (int j = lane; j < 1024; j += 32) {
            float e = __expf(row[j] - m);
            row[j] = e;
            sum += e;
        }
#pragma unroll
        for (int off = 16; off > 0; off >>= 1) sum += __shfl_xor(sum, off, 32);
        float inv = 1.f / sum;
        float* ga = a_out + ((size_t)bh * LT + t0 + r) * 1024;
        for (int j = lane; j < 1024; j += 32) {
            float p = row[j] * inv;
            row[j] = p;
            ga[j]  = p;
        }
    }
    __syncthreads();

    // ---- t_hat tile = P(16x1024) @ V(1024x128) + Q ; wave owns one d-tile
    {
        int d0 = wave * 16;
        v8f c = {};
        for (int kb = 0; kb < 256; ++kb) {
            int k = kb * 4;
            // A from LDS probabilities
            const float* pr = &S[(lane & 15) * 1024 + k + 2 * (lane >> 4)];
            v2f a; a.x = pr[0]; a.y = pr[1];
            // B = V[k][n]; V = concat(qt, qs) row-major. k,k+1 stay on one side
            // of the 512 boundary since k-block is 4-aligned.
            int kB = k + 2 * (lane >> 4);
            const float* vr = (kB < LT) ? (Kq + (size_t)kB * DQ)
                                        : (Ks + (size_t)(kB - LT) * DQ);
            int n = lane & 15;
            v2f b; b.x = vr[d0 + n]; b.y = vr[DQ + d0 + n];
            c = wmma4(a, b, c);
        }
        int b  = bh >> 3;
        int h  = bh & 7;
        int n  = lane & 15;
        int rb = 8 * (lane >> 4);
#pragma unroll
        for (int i = 0; i < 8; ++i) {
            int m = rb + i;
            float v = c[i] + Q[(size_t)m * DQ + d0 + n];     // + qt residual
            that[((size_t)(b * LT + t0 + m)) * NHD + h * DQ + d0 + n] = v;
        }
    }
}

// ---- Kernel 3: fc GEMM + residual + LayerNorm ----------------------------
// out = LN(that[16384x1024] @ W_fc[1024x128] + hla)
__global__ __launch_bounds__(256)
void fc_ln_kernel(const float* __restrict__ that, const float* __restrict__ Wfc,
                  const float* __restrict__ hla, float* __restrict__ out) {
    __shared__ float T[16 * 128];              // 8 KB
    int lane = threadIdx.x & 31;
    int wave = threadIdx.x >> 5;
    int r0   = blockIdx.x * 16;                // row block in [0, 16384)

    const float* A = that + (size_t)r0 * NHD;
    int d0 = wave * 16;
    v8f c = {};
    for (int kb = 0; kb < 256; ++kb) {
        v2f a = load_a_frag(A, NHD, kb * 4, lane);
        v2f b = load_b_kn(Wfc + d0, DMODEL, kb * 4, lane);
        c = wmma4(a, b, c);
    }
    int n  = lane & 15;
    int rb = 8 * (lane >> 4);
#pragma unroll
    for (int i = 0; i < 8; ++i) {
        int m = rb + i;
        T[m * DMODEL + d0 + n] = c[i] + hla[(size_t)(r0 + m) * DMODEL + d0 + n];
    }
    __syncthreads();

#pragma unroll
    for (int rr = 0; rr < 2; ++rr) {
        int r = wave * 2 + rr;
        const float* row = &T[r * DMODEL];
        float s = 0.f;
        for (int j = lane; j < DMODEL; j += 32) s += row[j];
#pragma unroll
        for (int off = 16; off > 0; off >>= 1) s += __shfl_xor(s, off, 32);
        float mu = s * (1.0f / DMODEL);
        float vs = 0.f;
        for (int j = lane; j < DMODEL; j += 32) {
            float d = row[j] - mu;
            vs += d * d;
        }
#pragma unroll
        for (int off = 16; off > 0; off >>= 1) vs += __shfl_xor(vs, off, 32);
        float rstd = rsqrtf(vs * (1.0f / DMODEL) + 1e-5f);
        float* go = out + (size_t)(r0 + r) * DMODEL;
        for (int j = lane; j < DMODEL; j += 32) go[j] = (row[j] - mu) * rstd;
    }
}

// ---------------------------------------------------------------------------

extern "C" void kernel_launch(void* const* d_in, const int* in_sizes, int n_in,
                              void* d_out, int out_size, void* d_ws, size_t ws_size,
                              hipStream_t stream) {
    const float* hla = (const float*)d_in[0];   // [32,512,128]
    const float* pep = (const float*)d_in[1];   // [32,512,128]
    const float* WQ  = (const float*)d_in[2];   // [128,1024]
    const float* WK  = (const float*)d_in[3];   // [128,1024]
    const float* Wfc = (const float*)d_in[4];   // [1024,128]

    float* outp  = (float*)d_out;
    float* aqq   = outp;                                   // 32*8*512*512
    float* aqs   = outp + 67108864ull;                     // 32*8*512*512
    float* a     = outp + 134217728ull;                    // 32*8*512*1024
    float* lnout = outp + 268435456ull;                    // 32*512*128

    float* qt   = (float*)d_ws;                            // 16,777,216 f
    float* qs   = qt + 16777216ull;                        // 16,777,216 f
    float* that = qs + 16777216ull;                        // 16,777,216 f

    // 1) projections: 131072 wave-tiles, 8 waves/block
    proj_kernel<<<16384, 256, 0, stream>>>(hla, pep, WQ, WK, qt, qs);

    // 2) fused scores + softmax + A*V (+ qt residual)
    dim3 g2(LT / 16, BS * NH);
    attn_kernel<<<g2, 256, 0, stream>>>(qt, qs, aqq, aqs, a, that);

    // 3) fc + residual + LayerNorm
    fc_ln_kernel<<<16384 / 16, 256, 0, stream>>>(that, Wfc, hla, lnout);
}